// SpikesEncoder_47991964565839
// MI455X (gfx1250) — compile-verified
//
#include <hip/hip_runtime.h>

typedef __attribute__((ext_vector_type(2))) float v2f;
typedef __attribute__((ext_vector_type(8))) float v8f;

#define N_UNITS 8192
#define N_STEPS 256
#define SPLITK  8
#define K_PER_SPLIT (N_UNITS / SPLITK)   // 1024

// ---------------------------------------------------------------------------
// Kernel 1: input_current = x @ W  via V_WMMA_F32_16X16X4_F32.
// Grid: 512 blocks (one per 16-column tile) x 256 threads (8 waves).
// Wave (tile, split) accumulates 1024 K-values in chunks of 4 per WMMA.
// All 16 rows of A carry the same x[k..k+3] chunk, so every row of C equals
// the partial matvec for columns j0..j0+15; lanes 0..15 store c[0].
// W is a 256 MB single-pass stream -> non-temporal loads (don't thrash the
// 192 MB L2; keep x resident instead). Each partial element is written
// exactly once -> no init, deterministic.
// ---------------------------------------------------------------------------
__global__ __launch_bounds__(256) void lif_matvec_wmma(
    const float* __restrict__ x,
    const float* __restrict__ W,
    float* __restrict__ partial)   // [SPLITK][N_UNITS]
{
    const int lane  = threadIdx.x & 31;
    const int wave  = threadIdx.x >> 5;     // 0..7
    const int tile  = blockIdx.x;           // 0..511
    const int split = wave;                 // split-K id
    const int j0    = tile * 16;
    const int n     = lane & 15;            // column within tile
    const int half  = lane >> 4;            // 0 -> K=0,1 ; 1 -> K=2,3
    const int kbase = split * K_PER_SPLIT;

    // A-layout (32-bit A 16x4): lanes 0-15 hold K=0 (vgpr0), K=1 (vgpr1);
    // lanes 16-31 hold K=2, K=3. Broadcast x chunk into all 16 rows.
    // B-layout (4x16): vgpr0 = row K=(0|2), vgpr1 = row K=(1|3), N = lane%16.
    const float* xp   = x + kbase + 2 * half;
    const float* wcol = W + (size_t)(kbase + 2 * half) * N_UNITS + j0 + n;

    v8f c = {};

#pragma unroll 8
    for (int k = 0; k < K_PER_SPLIT; k += 4) {
        v2f a, b;
        a.x = xp[0];                                   // x[k + 2*half]
        a.y = xp[1];                                   // x[k + 2*half + 1]
        b.x = __builtin_nontemporal_load(wcol);            // W[k+2h  ][j0+n]
        b.y = __builtin_nontemporal_load(wcol + N_UNITS);  // W[k+2h+1][j0+n]
        // (neg_a, A, neg_b, B, c_mod, C, reuse_a, reuse_b)
        c = __builtin_amdgcn_wmma_f32_16x16x4_f32(
                false, a, false, b, (short)0, c, false, false);
        xp   += 4;
        wcol += (size_t)4 * N_UNITS;
    }

    // C row 0: lanes 0-15 hold M=0, N=lane in vgpr0.
    if (lane < 16) {
        partial[split * N_UNITS + j0 + lane] = c[0];
    }
}

// ---------------------------------------------------------------------------
// Kernel 2: reduce split-K partials, then run the 256-step LIF scan.
// One thread per unit; spike stores are coalesced across the wave per step
// and written non-temporally (8 MB, never re-read).
//   V = (ALPHA*V + I) * (1 - Z);  Z = (V - 1 > 0);  out[t] = Z
// ---------------------------------------------------------------------------
__global__ __launch_bounds__(256) void lif_scan(
    const float* __restrict__ partial,   // [SPLITK][N_UNITS]
    float* __restrict__ out)             // [N_STEPS][N_UNITS]
{
    const int u = blockIdx.x * 256 + threadIdx.x;
    if (u >= N_UNITS) return;

    float I = 0.0f;
#pragma unroll
    for (int s = 0; s < SPLITK; ++s)
        I += partial[s * N_UNITS + u];

    const float ALPHA = (float)0.90483741803595952;  // exp(-dt/tau_m) = exp(-0.1)
    float V = 0.0f, Z = 0.0f;

#pragma unroll 8
    for (int t = 0; t < N_STEPS; ++t) {
        V = (ALPHA * V + I) * (1.0f - Z);
        Z = (V > 1.0f) ? 1.0f : 0.0f;
        __builtin_nontemporal_store(Z, &out[(size_t)t * N_UNITS + u]);
    }
}

// ---------------------------------------------------------------------------
extern "C" void kernel_launch(void* const* d_in, const int* in_sizes, int n_in,
                              void* d_out, int out_size, void* d_ws, size_t ws_size,
                              hipStream_t stream) {
    const float* x = (const float*)d_in[0];   // [8192]
    const float* W = (const float*)d_in[1];   // [8192, 8192]
    float* out     = (float*)d_out;           // [256, 8192]
    float* partial = (float*)d_ws;            // SPLITK * N_UNITS floats = 256 KB

    lif_matvec_wmma<<<N_UNITS / 16, 256, 0, stream>>>(x, W, partial);
    lif_scan<<<N_UNITS / 256, 256, 0, stream>>>(partial, out);
}